// TpsGridGen2_88682484728668
// MI455X (gfx1250) — compile-verified
//
#include <hip/hip_runtime.h>
#include <hip/hip_bf16.h>

typedef __attribute__((ext_vector_type(2))) float v2f;
typedef __attribute__((ext_vector_type(8))) float v8f;

#define TPS_N 9
#define TPS_B 32
#define OUT_H 512
#define OUT_W 512
#define WS_STRIDE 48   // floats per batch: PX[9] PY[9] coefX[12] coefY[12] pad

// ---------------------------------------------------------------------------
// Kernel 1: per-batch TPS solve (one thread per batch, 32 threads total).
// Solves [L | QX QY] (12x14) by Gauss-Jordan with partial pivoting.
// ws layout per batch: [0..8]=PX [9..17]=PY [18..29]=coefX(AX0,AX1,AX2,WX0..8)
//                      [30..41]=coefY
// ---------------------------------------------------------------------------
__global__ void tps_setup_kernel(const float* __restrict__ theta,
                                 float* __restrict__ ws) {
  int b = blockIdx.x * blockDim.x + threadIdx.x;
  if (b >= TPS_B) return;

  const float* th = theta + b * 4 * TPS_N;
  float QX[TPS_N], QY[TPS_N], PX[TPS_N], PY[TPS_N];
#pragma clang loop unroll(disable)
  for (int i = 0; i < TPS_N; ++i) {
    QX[i] = th[i];
    QY[i] = th[TPS_N + i];
    PX[i] = th[2 * TPS_N + i];
    PY[i] = th[3 * TPS_N + i];
  }

  float M[12][14];
#pragma clang loop unroll(disable)
  for (int i = 0; i < 12; ++i)
#pragma clang loop unroll(disable)
    for (int j = 0; j < 14; ++j) M[i][j] = 0.0f;

#pragma clang loop unroll(disable)
  for (int i = 0; i < TPS_N; ++i) {
#pragma clang loop unroll(disable)
    for (int j = 0; j < TPS_N; ++j) {
      float dx = PX[i] - PX[j];
      float dy = PY[i] - PY[j];
      float d2 = dx * dx + dy * dy;
      if (d2 == 0.0f) d2 = 1.0f;
      M[i][j] = d2 * __logf(d2);
    }
    M[i][9]  = 1.0f;  M[i][10] = PX[i]; M[i][11] = PY[i];
    M[9][i]  = 1.0f;  M[10][i] = PX[i]; M[11][i] = PY[i];
    M[i][12] = QX[i]; M[i][13] = QY[i];
  }

#pragma clang loop unroll(disable)
  for (int col = 0; col < 12; ++col) {
    int piv = col;
    float best = fabsf(M[col][col]);
#pragma clang loop unroll(disable)
    for (int r = col + 1; r < 12; ++r) {
      float v = fabsf(M[r][col]);
      if (v > best) { best = v; piv = r; }
    }
    if (piv != col) {
#pragma clang loop unroll(disable)
      for (int j = 0; j < 14; ++j) {
        float t = M[col][j]; M[col][j] = M[piv][j]; M[piv][j] = t;
      }
    }
    float inv = 1.0f / M[col][col];
#pragma clang loop unroll(disable)
    for (int j = 0; j < 14; ++j) M[col][j] *= inv;
#pragma clang loop unroll(disable)
    for (int r = 0; r < 12; ++r) {
      if (r == col) continue;
      float f = M[r][col];
#pragma clang loop unroll(disable)
      for (int j = 0; j < 14; ++j) M[r][j] -= f * M[col][j];
    }
  }

  float* o = ws + b * WS_STRIDE;
#pragma clang loop unroll(disable)
  for (int i = 0; i < TPS_N; ++i) { o[i] = PX[i]; o[9 + i] = PY[i]; }
  o[18] = M[9][12];  o[19] = M[10][12]; o[20] = M[11][12];
  o[30] = M[9][13];  o[31] = M[10][13]; o[32] = M[11][13];
#pragma clang loop unroll(disable)
  for (int i = 0; i < TPS_N; ++i) {
    o[21 + i] = M[i][12];
    o[33 + i] = M[i][13];
  }
}

// ---------------------------------------------------------------------------
// Kernel 2: streaming evaluation. One wave32 per 16-pixel tile; all 8 waves
// of a block share one (batch,row), derived from blockIdx only (scalar).
// NO local arrays anywhere: only named scalars, so ternaries lower to single
// v_cndmask instead of variable-index extract select-trees.
// K=12 contraction = three chained V_WMMA_F32_16X16X4_F32; D cols 0/1 are
// re-laid-out through LDS into one coalesced 128B store per wave.
// ---------------------------------------------------------------------------
__device__ __forceinline__ float tps_u(float gx, float gy, float px, float py) {
  float dx = gx - px, dy = gy - py;
  float d2 = dx * dx + dy * dy;
  return (d2 == 0.0f) ? 0.0f : d2 * __logf(d2);
}

__global__ __launch_bounds__(256) void tps_eval_kernel(
    const float* __restrict__ ws, float* __restrict__ out) {
  __shared__ float stage[8 * 32];

  const int lane = threadIdx.x & 31;
  const int wave = threadIdx.x >> 5;

  // Block-uniform decomposition: 4 blocks/row, 512 rows/batch, 32 batches.
  const int b   = blockIdx.x >> 11;
  const int h   = (blockIdx.x >> 2) & (OUT_H - 1);
  const int w16 = ((blockIdx.x & 3) << 3) + wave;

  const float* __restrict__ c = ws + b * WS_STRIDE;

  // Named scalar per-batch parameters (uniform -> SGPRs via s_load).
  const float px0 = c[0],  px1 = c[1],  px2 = c[2],  px3 = c[3],  px4 = c[4];
  const float px5 = c[5],  px6 = c[6],  px7 = c[7],  px8 = c[8];
  const float py0 = c[9],  py1 = c[10], py2 = c[11], py3 = c[12], py4 = c[13];
  const float py5 = c[14], py6 = c[15], py7 = c[16], py8 = c[17];
  const float cx0 = c[18], cx1 = c[19], cx2  = c[20], cx3  = c[21];
  const float cx4 = c[22], cx5 = c[23], cx6  = c[24], cx7  = c[25];
  const float cx8 = c[26], cx9 = c[27], cx10 = c[28], cx11 = c[29];
  const float cy0 = c[30], cy1 = c[31], cy2  = c[32], cy3  = c[33];
  const float cy4 = c[34], cy5 = c[35], cy6  = c[36], cy7  = c[37];
  const float cy8 = c[38], cy9 = c[39], cy10 = c[40], cy11 = c[41];

  const int  m  = lane & 15;          // pixel within tile / B,N column
  const bool kh = (lane >> 4) != 0;   // half-wave: K pair {2,3} vs {0,1}

  const int w = w16 * 16 + m;
  const float gx = -1.0f + 2.0f * (float)w * (1.0f / (float)(OUT_W - 1));
  const float gy = -1.0f + 2.0f * (float)h * (1.0f / (float)(OUT_H - 1));

  // Basis values (named scalars): [1, gx, gy, u0..u8]
  const float a0 = 1.0f, a1 = gx, a2 = gy;
  const float a3  = tps_u(gx, gy, px0, py0);
  const float a4  = tps_u(gx, gy, px1, py1);
  const float a5  = tps_u(gx, gy, px2, py2);
  const float a6  = tps_u(gx, gy, px3, py3);
  const float a7  = tps_u(gx, gy, px4, py4);
  const float a8  = tps_u(gx, gy, px5, py5);
  const float a9  = tps_u(gx, gy, px6, py6);
  const float a10 = tps_u(gx, gy, px7, py7);
  const float a11 = tps_u(gx, gy, px8, py8);

  const bool colX = (m == 0);
  const bool colY = (m == 1);

  v8f acc = {};

  // --- one K=4 step: A[m][k0..k3], B rows k0..k3 (cols 0/1 = x/y) ----------
#define TPS_WMMA_STEP(A0, A1, A2, A3, BX0, BX1, BX2, BX3, BY0, BY1, BY2, BY3) \
  {                                                                           \
    v2f a_frag, b_frag;                                                       \
    a_frag.x = kh ? (A2) : (A0);                                              \
    a_frag.y = kh ? (A3) : (A1);                                              \
    const float fx0 = kh ? (BX2) : (BX0);                                     \
    const float fx1 = kh ? (BX3) : (BX1);                                     \
    const float fy0 = kh ? (BY2) : (BY0);                                     \
    const float fy1 = kh ? (BY3) : (BY1);                                     \
    b_frag.x = colX ? fx0 : (colY ? fy0 : 0.0f);                              \
    b_frag.y = colX ? fx1 : (colY ? fy1 : 0.0f);                              \
    acc = __builtin_amdgcn_wmma_f32_16x16x4_f32(                              \
        false, a_frag, false, b_frag, (short)0, acc, false, false);           \
  }

  TPS_WMMA_STEP(a0, a1, a2,  a3,  cx0, cx1, cx2,  cx3,  cy0, cy1, cy2,  cy3)
  TPS_WMMA_STEP(a4, a5, a6,  a7,  cx4, cx5, cx6,  cx7,  cy4, cy5, cy6,  cy7)
  TPS_WMMA_STEP(a8, a9, a10, a11, cx8, cx9, cx10, cx11, cy8, cy9, cy10, cy11)
#undef TPS_WMMA_STEP

  // D layout: VGPR v -> lanes 0-15: M=v,N=lane ; lanes 16-31: M=v+8,N=lane-16.
  // Meaningful columns: N=0 (xp) on lanes 0/16, N=1 (yp) on lanes 1/17.
  float* tl = stage + wave * 32;
  if (m < 2) {
    const int rbase = kh ? 8 : 0;
    tl[(0 + rbase) * 2 + m] = acc[0];
    tl[(1 + rbase) * 2 + m] = acc[1];
    tl[(2 + rbase) * 2 + m] = acc[2];
    tl[(3 + rbase) * 2 + m] = acc[3];
    tl[(4 + rbase) * 2 + m] = acc[4];
    tl[(5 + rbase) * 2 + m] = acc[5];
    tl[(6 + rbase) * 2 + m] = acc[6];
    tl[(7 + rbase) * 2 + m] = acc[7];
  }
  __syncthreads();

  // One coalesced 128B store per wave: 16 pixels * {x,y} interleaved.
  const size_t base =
      (((size_t)b * OUT_H + (size_t)h) * OUT_W + (size_t)(w16 * 16)) * 2;
  out[base + lane] = tl[lane];
}

extern "C" void kernel_launch(void* const* d_in, const int* in_sizes, int n_in,
                              void* d_out, int out_size, void* d_ws, size_t ws_size,
                              hipStream_t stream) {
  const float* theta = (const float*)d_in[0];
  float* ws  = (float*)d_ws;
  float* out = (float*)d_out;

  tps_setup_kernel<<<1, 32, 0, stream>>>(theta, ws);
  tps_eval_kernel<<<65536, 256, 0, stream>>>(ws, out);
}